// WaterNet3_76373108457553
// MI455X (gfx1250) — compile-verified
//
#include <hip/hip_runtime.h>
#include <hip/hip_bf16.h>
#include <math.h>

#define NT 1000
#define NS 512
#define NH 64
#define NG 16

typedef __attribute__((ext_vector_type(2))) float v2f;
typedef __attribute__((ext_vector_type(8))) float v8f;

__device__ __forceinline__ float sigm(float x) {
    return 1.0f / (1.0f + __expf(-x));
}

// ---------------------------------------------------------------------------
// Setup kernel: one wave (32 threads) per 16-site group.
// Computes w = xc @ W_fc^T + b_fc via chained V_WMMA_F32_16X16X4_F32 (K=16 as
// 4 slices of 4), applies the per-segment nonlinearity, and computes
// base = xc @ W_fcT[:,1:]^T + b_fcT the same way.
// params layout (floats): [seg][s*NH + h], seg 0..7 = gm,ge,k2,k23,k3,gl,
// ga_logit(raw),qb ; seg 8 = base.
// ---------------------------------------------------------------------------
__global__ __launch_bounds__(32) void waternet_setup_wmma(
    const float* __restrict__ xc,     // [NS, NG]
    const float* __restrict__ W_fc,   // [8*NH+1, NG]
    const float* __restrict__ b_fc,   // [8*NH+1]
    const float* __restrict__ W_fcT,  // [NH, 1+NG]
    const float* __restrict__ b_fcT,  // [NH]
    float* __restrict__ params)
{
    const int lane = threadIdx.x & 31;
    const int half = lane >> 4;       // 0: lanes 0-15, 1: lanes 16-31
    const int m    = lane & 15;
    const int koff = 2 * half;        // K offset within 4-slice
    const int s0   = blockIdx.x * 16; // 16 sites per block

    // A fragments: xc tile [16 sites x 16 (K)] -> 4 fragments of K=4
    v2f aF[4];
#pragma unroll
    for (int kk = 0; kk < 4; ++kk) {
        const float* ap = xc + (s0 + m) * NG + 4 * kk + koff;
        aF[kk].x = ap[0];
        aF[kk].y = ap[1];
    }

    // ---- w tiles: 32 tiles cover output columns 0..511 (col 512 unused) ----
    for (int jt = 0; jt < 32; ++jt) {
        const int j0 = jt * 16;
        v2f bF[4];
#pragma unroll
        for (int kk = 0; kk < 4; ++kk) {
            // B[k, n] = W_fc^T[k, j0+n] = W_fc[(j0+n)*NG + k]
            const float* bp = W_fc + (j0 + m) * NG + 4 * kk + koff;
            bF[kk].x = bp[0];
            bF[kk].y = bp[1];
        }
        v8f c = {};
#pragma unroll
        for (int kk = 0; kk < 4; ++kk) {
            c = __builtin_amdgcn_wmma_f32_16x16x4_f32(
                false, aF[kk], false, bF[kk], (short)0, c, false, false);
        }
        const float bias = b_fc[j0 + m];
        const int   seg  = j0 >> 6;          // tile fully inside one segment
        const int   h    = (j0 & 63) + m;    // hidden index within segment
        float* dst = params + (size_t)seg * (NS * NH);
#pragma unroll
        for (int r = 0; r < 8; ++r) {
            const int   s   = s0 + r + 8 * half; // C layout: M = r + 8*half
            const float val = c[r] + bias;
            float out;
            switch (seg) {
                case 0:  out = __expf(val) + 1.0f;  break; // gm
                case 1:  out = 2.0f * sigm(val);    break; // ge
                case 2:  out = sigm(val);           break; // k2
                case 3:  out = sigm(val);           break; // k23
                case 4:  out = 0.1f * sigm(val);    break; // k3
                case 5:  out = 2.0f * __expf(val);  break; // gl
                case 6:  out = val;                 break; // ga raw logit
                default: out = fmaxf(val, 0.0f);    break; // qb
            }
            dst[s * NH + h] = out;
        }
    }

    // ---- base tiles: 4 tiles cover hidden cols 0..63 ----
    for (int jt = 0; jt < 4; ++jt) {
        const int h0 = jt * 16;
        v2f bF[4];
#pragma unroll
        for (int kk = 0; kk < 4; ++kk) {
            // B[k, n] = W_fcT[(h0+n), 1+k]
            const float* bp = W_fcT + (h0 + m) * (1 + NG) + 1 + 4 * kk + koff;
            bF[kk].x = bp[0];
            bF[kk].y = bp[1];
        }
        v8f c = {};
#pragma unroll
        for (int kk = 0; kk < 4; ++kk) {
            c = __builtin_amdgcn_wmma_f32_16x16x4_f32(
                false, aF[kk], false, bF[kk], (short)0, c, false, false);
        }
        const float bias = b_fcT[h0 + m];
        float* dst = params + (size_t)8 * (NS * NH);
#pragma unroll
        for (int r = 0; r < 8; ++r) {
            const int s = s0 + r + 8 * half;
            dst[s * NH + h0 + m] = c[r] + bias;
        }
    }
}

// ---------------------------------------------------------------------------
// Scan kernel: one wave32 per site; each lane owns hidden units (lane, lane+32).
// All recurrent state in VGPRs; per-step output reduction via __shfl_xor tree.
// ---------------------------------------------------------------------------
__global__ __launch_bounds__(256) void waternet_scan(
    const float* __restrict__ x,      // [NT, NS, 5]
    const float* __restrict__ W_fcT,  // [NH, 1+NG]
    const float* __restrict__ params, // [9][NS*NH]
    float* __restrict__ Y)            // [NT, NS]
{
    const int lane = threadIdx.x & 31;
    const int wid  = threadIdx.x >> 5;
    const int s    = blockIdx.x * 8 + wid;
    const int h0   = lane;
    const int h1   = lane + 32;

    const size_t SN = (size_t)NS * NH;
    const size_t b0 = (size_t)s * NH + h0;
    const size_t b1 = (size_t)s * NH + h1;

    // Time-invariant per-(s,h) parameters -> registers
    const float gm0   = params[0 * SN + b0], gm1   = params[0 * SN + b1];
    const float ge0   = params[1 * SN + b0], ge1   = params[1 * SN + b1];
    const float k20   = params[2 * SN + b0], k21   = params[2 * SN + b1];
    const float k230  = params[3 * SN + b0], k231  = params[3 * SN + b1];
    const float k30   = params[4 * SN + b0], k31   = params[4 * SN + b1];
    const float gl0   = params[5 * SN + b0], gl1   = params[5 * SN + b1];
    const float lg0   = params[6 * SN + b0], lg1   = params[6 * SN + b1];
    const float qb0   = params[7 * SN + b0], qb1   = params[7 * SN + b1];
    const float base0 = params[8 * SN + b0], base1 = params[8 * SN + b1];
    const float omk230 = 1.0f - k230, omk231 = 1.0f - k231;

    // Softmax of ga logits over the 64 hidden units of this site (intra-wave)
    float mx = fmaxf(lg0, lg1);
#pragma unroll
    for (int off = 16; off; off >>= 1)
        mx = fmaxf(mx, __shfl_xor(mx, off, 32));
    float e0 = __expf(lg0 - mx), e1 = __expf(lg1 - mx);
    float ssum = e0 + e1;
#pragma unroll
    for (int off = 16; off; off >>= 1)
        ssum += __shfl_xor(ssum, off, 32);
    const float inv = 1.0f / ssum;
    const float ga0 = e0 * inv, ga1 = e1 * inv;

    // LAI weight (column 0 of W_fcT)
    const float w00 = W_fcT[h0 * (1 + NG)];
    const float w01 = W_fcT[h1 * (1 + NG)];

    float S00 = 0.0f, S20 = 0.0f, S30 = 0.0f;
    float S01 = 0.0f, S21 = 0.0f, S31 = 0.0f;

    const float invPI = 1.0f / 3.1415f;

    for (int t = 0; t < NT; ++t) {
        const float* xp = x + ((size_t)t * NS + s) * 5;
        const float Pp  = xp[0];
        const float E   = xp[1];
        const float T1  = xp[2];
        const float T2  = xp[3];
        const float LAI = xp[4];
        if (t + 1 < NT)
            __builtin_prefetch(xp + (size_t)NS * 5, 0, 0); // global_prefetch_b8

        const float Ta = 0.5f * (T1 + T2);
        const bool  valid = (T1 < 0.0f) && (T2 > 0.0f);
        const float arg = valid ? (T1 + T2) / (T2 - T1) : 0.0f;
        float rP = 1.0f - acosf(arg) * invPI;
        if (T1 >= 0.0f) rP = 1.0f;
        if (T2 <= 0.0f) rP = 0.0f;           // T2<=0 mask applied last -> wins
        const float Ps = (1.0f - rP) * Pp;
        const float Pl = rP * Pp;

        // ---- hidden unit h0 ----
        const float vi0  = sigm(fmaf(LAI, w00, base0));
        const float H0a  = S00 + Ps;
        const float qSm0 = fminf(H0a, fmaxf(Ta * gm0, 0.0f));
        const float qIn0 = qSm0 + Pl * vi0 - E * ge0;
        const float t2a0 = S20 + qIn0;
        const float Q10  = fmaxf(t2a0 - gl0, 0.0f);
        const float H2a  = fminf(fmaxf(t2a0, 0.0f), gl0);
        const float q2a  = H2a * k20;
        const float Q20  = q2a * omk230;
        const float H3a  = S30 + q2a * k230;
        const float Q30  = fmaxf(fmaf(H3a, k30, qb0), 0.0f);
        float acc = (Q10 + Q20 + Q30) * ga0;
        S00 = H0a - qSm0; S20 = H2a - Q20; S30 = H3a - Q30;

        // ---- hidden unit h1 ----
        const float vi1  = sigm(fmaf(LAI, w01, base1));
        const float H0b  = S01 + Ps;
        const float qSm1 = fminf(H0b, fmaxf(Ta * gm1, 0.0f));
        const float qIn1 = qSm1 + Pl * vi1 - E * ge1;
        const float t2b1 = S21 + qIn1;
        const float Q11  = fmaxf(t2b1 - gl1, 0.0f);
        const float H2b  = fminf(fmaxf(t2b1, 0.0f), gl1);
        const float q2b  = H2b * k21;
        const float Q21  = q2b * omk231;
        const float H3b  = S31 + q2b * k231;
        const float Q31  = fmaxf(fmaf(H3b, k31, qb1), 0.0f);
        acc += (Q11 + Q21 + Q31) * ga1;
        S01 = H0b - qSm1; S21 = H2b - Q21; S31 = H3b - Q31;

        // Reduce 64 hidden units (2 per lane already summed) across the wave
        float sum = acc;
#pragma unroll
        for (int off = 16; off; off >>= 1)
            sum += __shfl_xor(sum, off, 32);
        if (lane == 0)
            Y[(size_t)t * NS + s] = sum;
    }
}

extern "C" void kernel_launch(void* const* d_in, const int* in_sizes, int n_in,
                              void* d_out, int out_size, void* d_ws, size_t ws_size,
                              hipStream_t stream) {
    (void)in_sizes; (void)n_in; (void)out_size; (void)ws_size;
    const float* x     = (const float*)d_in[0];  // [NT, NS, 5]
    const float* xc    = (const float*)d_in[1];  // [NS, NG]
    const float* W_fc  = (const float*)d_in[2];  // [8*NH+1, NG]
    const float* b_fc  = (const float*)d_in[3];  // [8*NH+1]
    const float* W_fcT = (const float*)d_in[4];  // [NH, 1+NG]
    const float* b_fcT = (const float*)d_in[5];  // [NH]
    float* params = (float*)d_ws;                // 9 * NS * NH floats (~1.2 MB)
    float* Y      = (float*)d_out;               // [NT, NS]

    waternet_setup_wmma<<<NS / 16, 32, 0, stream>>>(xc, W_fc, b_fc, W_fcT,
                                                    b_fcT, params);
    waternet_scan<<<NS / 8, 256, 0, stream>>>(x, W_fcT, params, Y);
}